// Net_17789754541039
// MI455X (gfx1250) — compile-verified
//
#include <hip/hip_runtime.h>
#include <math.h>

typedef float v2f __attribute__((ext_vector_type(2)));
typedef float v8f __attribute__((ext_vector_type(8)));

// ---------------------------------------------------------------------------
// Scatter-add of messages: agg[dst] += feat[src] * ew   (CPE = channels/node)
// CPE/16 threads cooperate on one edge; each does 4 float4 gathers + 16 atomics.
// agg region lives in L2 (<=25.6MB), so atomics resolve at L2 speed.
// ---------------------------------------------------------------------------
template <int CPE>
__global__ void scatter_kernel(const float* __restrict__ feat,
                               const int* __restrict__ src,
                               const int* __restrict__ dst,
                               const float* __restrict__ ew,
                               float* __restrict__ agg,
                               int E)
{
    constexpr int CHUNKS = CPE / 16;              // threads per edge
    int t = blockIdx.x * blockDim.x + threadIdx.x;
    int e = t / CHUNKS;
    if (e >= E) return;
    int c0 = (t % CHUNKS) * 16;

    int s = src[e];
    int d = dst[e];
    float w = ew[e];

    const float4* xs = (const float4*)(feat + (size_t)s * CPE + c0);
    float* ag = agg + (size_t)d * CPE + c0;
#pragma unroll
    for (int q = 0; q < 4; ++q) {
        float4 v = xs[q];
        atomicAdd(ag + q * 4 + 0, v.x * w);
        atomicAdd(ag + q * 4 + 1, v.y * w);
        atomicAdd(ag + q * 4 + 2, v.z * w);
        atomicAdd(ag + q * 4 + 3, v.w * w);
    }
}

// ---------------------------------------------------------------------------
// Layer 1 fused dual-GEMM via V_WMMA_F32_16X16X4_F32:
//   h1 = relu(agg @ W_rel^T + x @ W_root^T + b)   [N,64]x[64,32] -> [N,32]
// One wave per 16-row node tile; 2 column tiles of 16; K=64 in steps of 4.
// A layout (16x4 f32): lane m=lane&15, vgpr0:k=k0+2*hi, vgpr1:k=k0+1+2*hi.
// B layout (4x16 f32): lane n=lane&15, same k striping.
// D layout (v8f):      row = tile*16 + i + 8*hi, col = colBase + (lane&15).
// ---------------------------------------------------------------------------
__global__ void gemm1_kernel(const float* __restrict__ agg,    // [N,64]
                             const float* __restrict__ x,      // [N,64]
                             const float* __restrict__ Wrel,   // [32,64] row-major
                             const float* __restrict__ Wroot,  // [32,64]
                             const float* __restrict__ bias,   // [32]
                             float* __restrict__ h1,           // [N,32]
                             int N, int nTiles)
{
    const int lane = threadIdx.x & 31;
    const int tile = blockIdx.x * (blockDim.x >> 5) + (threadIdx.x >> 5);
    if (tile >= nTiles) return;                   // wave-uniform exit

    const int row0 = tile * 16;
    const int hi   = lane >> 4;                   // 0: k pair {0,1}, 1: {2,3}
    const int m    = lane & 15;
    const int aRow = min(row0 + m, N - 1);        // clamped load row (no divergence)

    for (int ct = 0; ct < 2; ++ct) {
        const int colBase = ct * 16;
        const int n = colBase + m;                // B / output column
        v8f acc = {0.f, 0.f, 0.f, 0.f, 0.f, 0.f, 0.f, 0.f};
#pragma unroll
        for (int k0 = 0; k0 < 64; k0 += 4) {
            const int kb = k0 + hi * 2;           // even -> 8B aligned v2f loads
            v2f aRel  = *(const v2f*)(agg   + (size_t)aRow * 64 + kb);
            v2f bRel  = *(const v2f*)(Wrel  + (size_t)n    * 64 + kb);
            acc = __builtin_amdgcn_wmma_f32_16x16x4_f32(false, aRel, false, bRel,
                                                        (short)0, acc, false, false);
            v2f aRoot = *(const v2f*)(x     + (size_t)aRow * 64 + kb);
            v2f bRoot = *(const v2f*)(Wroot + (size_t)n    * 64 + kb);
            acc = __builtin_amdgcn_wmma_f32_16x16x4_f32(false, aRoot, false, bRoot,
                                                        (short)0, acc, false, false);
        }
        const float bn = bias[n];
#pragma unroll
        for (int i = 0; i < 8; ++i) {
            int r = row0 + i + 8 * hi;
            if (r < N)                            // predicated stores, after all WMMA
                h1[(size_t)r * 32 + n] = fmaxf(acc[i] + bn, 0.0f);
        }
    }
}

// ---------------------------------------------------------------------------
// Layer 2 fused dual-GEMM via WMMA: g = agg2 @ W2_rel^T + h1 @ W2_root^T + b2
// [N,32]x[32,8] -> [N,8]. Output cols 8..15 of the 16-wide tile use clamped
// weight rows and are discarded (never stored).
// ---------------------------------------------------------------------------
__global__ void gemm2_kernel(const float* __restrict__ agg2,   // [N,32]
                             const float* __restrict__ h1,     // [N,32]
                             const float* __restrict__ Wrel,   // [8,32]
                             const float* __restrict__ Wroot,  // [8,32]
                             const float* __restrict__ bias,   // [8]
                             float* __restrict__ g,            // [N,8]
                             int N, int nTiles)
{
    const int lane = threadIdx.x & 31;
    const int tile = blockIdx.x * (blockDim.x >> 5) + (threadIdx.x >> 5);
    if (tile >= nTiles) return;

    const int row0 = tile * 16;
    const int hi   = lane >> 4;
    const int m    = lane & 15;
    const int aRow = min(row0 + m, N - 1);
    const int nc   = (m < 8) ? m : 0;             // clamped weight row (cols 8..15 dup col 0)

    v8f acc = {0.f, 0.f, 0.f, 0.f, 0.f, 0.f, 0.f, 0.f};
#pragma unroll
    for (int k0 = 0; k0 < 32; k0 += 4) {
        const int kb = k0 + hi * 2;
        v2f aRel  = *(const v2f*)(agg2  + (size_t)aRow * 32 + kb);
        v2f bRel  = *(const v2f*)(Wrel  + (size_t)nc   * 32 + kb);
        acc = __builtin_amdgcn_wmma_f32_16x16x4_f32(false, aRel, false, bRel,
                                                    (short)0, acc, false, false);
        v2f aRoot = *(const v2f*)(h1    + (size_t)aRow * 32 + kb);
        v2f bRoot = *(const v2f*)(Wroot + (size_t)nc   * 32 + kb);
        acc = __builtin_amdgcn_wmma_f32_16x16x4_f32(false, aRoot, false, bRoot,
                                                    (short)0, acc, false, false);
    }
    if (m < 8) {                                  // divergence only after all WMMA
        const float bn = bias[m];
#pragma unroll
        for (int i = 0; i < 8; ++i) {
            int r = row0 + i + 8 * hi;
            if (r < N)
                g[(size_t)r * 8 + m] = acc[i] + bn;
        }
    }
}

// ---------------------------------------------------------------------------
// log_softmax over 8 channels + fused head: out = relu([emb, x1] @ W_lin^T + b)
// d_out layout: out[N] then embedding[N*8] (tuple return order, flattened).
// ---------------------------------------------------------------------------
__global__ void finalize_kernel(const float* __restrict__ g,     // [N,8]
                                const float* __restrict__ x1,    // [N]
                                const float* __restrict__ Wlin,  // [9]
                                const float* __restrict__ blin,  // [1]
                                float* __restrict__ out,         // [9N]
                                int N)
{
    int i = blockIdx.x * blockDim.x + threadIdx.x;
    if (i >= N) return;

    const float4* gp = (const float4*)(g + (size_t)i * 8);
    float4 g0 = gp[0], g1 = gp[1];
    float v[8] = {g0.x, g0.y, g0.z, g0.w, g1.x, g1.y, g1.z, g1.w};

    float mx = v[0];
#pragma unroll
    for (int j = 1; j < 8; ++j) mx = fmaxf(mx, v[j]);
    float s = 0.0f;
#pragma unroll
    for (int j = 0; j < 8; ++j) s += __expf(v[j] - mx);
    float lse = __logf(s);

    float acc = blin[0] + x1[i] * Wlin[8];
    float* emb = out + N + (size_t)i * 8;
#pragma unroll
    for (int j = 0; j < 8; ++j) {
        float e = v[j] - mx - lse;
        emb[j] = e;
        acc += e * Wlin[j];
    }
    out[i] = fmaxf(acc, 0.0f);
}

// ---------------------------------------------------------------------------
extern "C" void kernel_launch(void* const* d_in, const int* in_sizes, int n_in,
                              void* d_out, int out_size, void* d_ws, size_t ws_size,
                              hipStream_t stream)
{
    const float* x    = (const float*)d_in[0];   // [N,64]
    const int*   ei   = (const int*)  d_in[1];   // [2,E]
    const float* ew   = (const float*)d_in[2];   // [E]
    const float* x1   = (const float*)d_in[3];   // [N,1]
    const float* W1r  = (const float*)d_in[4];   // [32,64]
    const float* b1   = (const float*)d_in[5];   // [32]
    const float* W1o  = (const float*)d_in[6];   // [32,64]
    const float* W2r  = (const float*)d_in[7];   // [8,32]
    const float* b2   = (const float*)d_in[8];   // [8]
    const float* W2o  = (const float*)d_in[9];   // [8,32]
    const float* Wlin = (const float*)d_in[10];  // [1,9]
    const float* blin = (const float*)d_in[11];  // [1]

    const int N = in_sizes[0] / 64;
    const int E = in_sizes[2];
    const int* src = ei;
    const int* dst = ei + E;

    // Workspace layout (floats): agg1[N*64] | agg2[N*32] | h1[N*32] | g[N*8]
    float* agg1 = (float*)d_ws;
    float* agg2 = agg1 + (size_t)N * 64;
    float* h1   = agg2 + (size_t)N * 32;
    float* g    = h1   + (size_t)N * 32;

    // Zero both atomic accumulator regions (contiguous: N*96 floats).
    hipMemsetAsync(agg1, 0, (size_t)N * 96 * sizeof(float), stream);

    const int BLK = 256;

    // Layer 1 scatter: 4 threads per edge.
    {
        long long work = (long long)E * 4;
        int blocks = (int)((work + BLK - 1) / BLK);
        scatter_kernel<64><<<blocks, BLK, 0, stream>>>(x, src, dst, ew, agg1, E);
    }

    // Layer 1 GEMM (WMMA) + bias + relu.
    const int nTiles = (N + 15) / 16;
    {
        int wavesPerBlock = BLK / 32;
        int blocks = (nTiles + wavesPerBlock - 1) / wavesPerBlock;
        gemm1_kernel<<<blocks, BLK, 0, stream>>>(agg1, x, W1r, W1o, b1, h1, N, nTiles);
    }

    // Layer 2 scatter: 2 threads per edge.
    {
        long long work = (long long)E * 2;
        int blocks = (int)((work + BLK - 1) / BLK);
        scatter_kernel<32><<<blocks, BLK, 0, stream>>>(h1, src, dst, ew, agg2, E);
    }

    // Layer 2 GEMM (WMMA) + bias.
    {
        int wavesPerBlock = BLK / 32;
        int blocks = (nTiles + wavesPerBlock - 1) / wavesPerBlock;
        gemm2_kernel<<<blocks, BLK, 0, stream>>>(agg2, h1, W2r, W2o, b2, g, N, nTiles);
    }

    // log_softmax + linear head, write both outputs.
    {
        int blocks = (N + BLK - 1) / BLK;
        finalize_kernel<<<blocks, BLK, 0, stream>>>(g, x1, Wlin, blin, (float*)d_out, N);
    }
}